// Particles2Grid_18915035972383
// MI455X (gfx1250) — compile-verified
//
#include <hip/hip_runtime.h>

typedef __attribute__((ext_vector_type(4))) unsigned int v4u;
typedef __attribute__((ext_vector_type(8))) int v8i;
typedef __attribute__((ext_vector_type(4))) int v4i;

#define GRID_N   128
#define NCELLS   (128 * 128 * 128)
#define CHUNK    256
#define SIGMA    318.30988618379067f   // 8 / (pi * h^3), h = 0.2
#define H2       0.04f                 // h^2
#define INV_H    5.0f                  // 1/h
#define INV_STEP 10.0f                 // 1/0.1
#define STEP     0.1f

// ---------------------------------------------------------------------------
// 1-D TDM load: copy n_elems f32 elements from global memory into LDS via the
// Tensor Data Mover (gfx1250). Issued once by wave 0; tracked by TENSORcnt.
// 6-arg builtin variant (clang-23 / therock-10.0 headers).
// ---------------------------------------------------------------------------
__device__ __forceinline__ void tdm_load_1d(unsigned int lds_addr,
                                            const void* gptr,
                                            unsigned int n_elems) {
  unsigned long long ga = (unsigned long long)(uintptr_t)gptr;

  v4u g0;
  g0.x = 1u;                                   // count=1, user mode, no gather
  g0.y = lds_addr;                             // LDS byte address
  g0.z = (unsigned int)ga;                     // global_addr[31:0]
  g0.w = (unsigned int)((ga >> 32) & 0x01FFFFFFu) | (2u << 30); // addr[56:32], type=2

  v8i g1;
  g1[0] = (int)(2u << 16);                     // workgroup_mask=0, data_size=4B
  g1[1] = (int)((n_elems & 0xFFFFu) << 16);    // tensor_dim0 lo16 in [63:48]
  g1[2] = (int)(((n_elems >> 16) & 0xFFFFu)    // tensor_dim0 hi16
                | (1u << 16));                 // tensor_dim1 = 1 (lo16)
  g1[3] = (int)((n_elems & 0xFFFFu) << 16);    // tensor_dim1 hi=0, tile_dim0=n
  g1[4] = 1;                                   // tile_dim1=1, tile_dim2=0
  g1[5] = (int)n_elems;                        // tensor_dim0_stride lo32
  g1[6] = 0;                                   // stride0 hi, stride1 lo
  g1[7] = 0;                                   // stride1 hi

  v4i z4 = {0, 0, 0, 0};
  v8i z8 = {0, 0, 0, 0, 0, 0, 0, 0};
  __builtin_amdgcn_tensor_load_to_lds(g0, g1, z4, z4, z8, 0);
}

// ---------------------------------------------------------------------------
// Zero-fill output (must run every launch: graph replays accumulate).
// ---------------------------------------------------------------------------
__global__ void p2g_zero_kernel(float4* __restrict__ out, int n4) {
  int i = blockIdx.x * blockDim.x + threadIdx.x;
  if (i < n4) out[i] = float4{0.0f, 0.0f, 0.0f, 0.0f};
}

// ---------------------------------------------------------------------------
// Particle->grid splat. One particle per lane; TDM stages the block's chunk
// of locs/data into LDS; scatter via native f32 global atomics (agent scope).
// Outer loops kept rolled (code-size / I$), inner z-loop unrolled.
// ---------------------------------------------------------------------------
__global__ void p2g_splat_kernel(const float4* __restrict__ locs,
                                 const float4* __restrict__ data,
                                 const float*  __restrict__ density,
                                 float* __restrict__ out, int N) {
  __shared__ float4 sLocs[CHUNK];
  __shared__ float4 sData[CHUNK];

  const int b      = blockIdx.y;
  const int base   = blockIdx.x * CHUNK;
  const int chunkN = min(CHUNK, N - base);

  // Wave 0 issues the async tensor DMA and waits for completion; the
  // workgroup barrier then makes the LDS tiles visible to all 8 waves.
  if (threadIdx.x < 32) {
    unsigned int nEl = (unsigned int)chunkN * 4u;  // f32 elements
    tdm_load_1d((unsigned int)(uintptr_t)&sLocs[0],
                locs + (size_t)b * N + base, nEl);
    tdm_load_1d((unsigned int)(uintptr_t)&sData[0],
                data + (size_t)b * N + base, nEl);
    __builtin_amdgcn_s_wait_tensorcnt(0);
  }
  __syncthreads();

  const int t = threadIdx.x;
  if (t >= chunkN) return;

  const float4 L = sLocs[t];                 // pos.xyz, inv_mass
  const float4 D = sData[t];
  const float dens = density[(size_t)b * N + base + t];
  const float cs = SIGMA / (L.w * dens);     // sigma / (inv_mass * density)

  const int bx = (int)floorf(L.x * INV_STEP);
  const int by = (int)floorf(L.y * INV_STEP);
  const int bz = (int)floorf(L.z * INV_STEP);

  float* __restrict__ outB = out + (size_t)b * ((size_t)NCELLS * 4);

  #pragma unroll 1
  for (int ox = -2; ox <= 2; ++ox) {
    const int cx = bx + ox;
    if ((unsigned)cx >= (unsigned)GRID_N) continue;
    const float fx = ((float)cx + 0.5f) * STEP - L.x;
    const float dx2 = fx * fx;
    if (dx2 >= H2) continue;                 // whole y/z slab outside sphere
    #pragma unroll 1
    for (int oy = -2; oy <= 2; ++oy) {
      const int cy = by + oy;
      if ((unsigned)cy >= (unsigned)GRID_N) continue;
      const float fy = ((float)cy + 0.5f) * STEP - L.y;
      const float dxy2 = dx2 + fy * fy;
      if (dxy2 >= H2) continue;              // whole z row outside sphere
      const int rowBase = (cx * GRID_N + cy) * GRID_N;
      #pragma unroll
      for (int oz = -2; oz <= 2; ++oz) {
        const int cz = bz + oz;
        if ((unsigned)cz >= (unsigned)GRID_N) continue;
        const float fz = ((float)cz + 0.5f) * STEP - L.z;
        const float d2 = dxy2 + fz * fz;
        if (d2 >= H2) continue;              // w == 0: skip 4 atomics
        const float q = sqrtf(d2) * INV_H;
        float w;
        if (q < 0.5f) {
          w = 1.0f + 6.0f * q * q * (q - 1.0f);   // 1 - 6q^2 + 6q^3
        } else {
          const float u = 1.0f - q;
          w = 2.0f * u * u * u;
        }
        const float coef = w * cs;
        float* p = outB + ((size_t)(rowBase + cz) << 2);
        // relaxed agent-scope f32 adds -> native global_atomic_add_f32 (L2)
        (void)__hip_atomic_fetch_add(p + 0, coef * D.x, __ATOMIC_RELAXED,
                                     __HIP_MEMORY_SCOPE_AGENT);
        (void)__hip_atomic_fetch_add(p + 1, coef * D.y, __ATOMIC_RELAXED,
                                     __HIP_MEMORY_SCOPE_AGENT);
        (void)__hip_atomic_fetch_add(p + 2, coef * D.z, __ATOMIC_RELAXED,
                                     __HIP_MEMORY_SCOPE_AGENT);
        (void)__hip_atomic_fetch_add(p + 3, coef * D.w, __ATOMIC_RELAXED,
                                     __HIP_MEMORY_SCOPE_AGENT);
      }
    }
  }
}

extern "C" void kernel_launch(void* const* d_in, const int* in_sizes, int n_in,
                              void* d_out, int out_size, void* d_ws, size_t ws_size,
                              hipStream_t stream) {
  const float* locs    = (const float*)d_in[0];   // (B, N, 4) f32
  const float* data    = (const float*)d_in[1];   // (B, N, 4) f32
  const float* density = (const float*)d_in[2];   // (B, N)    f32

  const int B = out_size / (NCELLS * 4);          // (B,128,128,128,4)
  if (B <= 0) return;
  const int N = in_sizes[2] / B;

  // 1) zero the accumulator grid (vectorized float4 stores)
  const int n4 = out_size / 4;
  p2g_zero_kernel<<<(n4 + 255) / 256, 256, 0, stream>>>((float4*)d_out, n4);

  // 2) scatter-add splat
  dim3 grid((N + CHUNK - 1) / CHUNK, B);
  p2g_splat_kernel<<<grid, CHUNK, 0, stream>>>(
      (const float4*)locs, (const float4*)data, density, (float*)d_out, N);
}